// AttnReweight_85117661872427
// MI455X (gfx1250) — compile-verified
//
#include <hip/hip_runtime.h>
#include <math.h>

#define B_   2
#define HD_  8
#define H_   192
#define W_   192
#define K_   9
#define NSP_ 9

typedef __attribute__((ext_vector_type(2))) float v2f;
typedef __attribute__((ext_vector_type(8))) float v8f;

// ---------------- global max reduction (for c = max(attn)) ----------------
__global__ __launch_bounds__(256) void reduce_max_stage1(
    const float* __restrict__ x, int n, float* __restrict__ partial)
{
    __shared__ float sm[256];
    float m = -INFINITY;
    for (int i = blockIdx.x * 256 + threadIdx.x; i < n; i += gridDim.x * 256)
        m = fmaxf(m, x[i]);
    sm[threadIdx.x] = m;
    __syncthreads();
    for (int s = 128; s > 0; s >>= 1) {
        if (threadIdx.x < s) sm[threadIdx.x] = fmaxf(sm[threadIdx.x], sm[threadIdx.x + s]);
        __syncthreads();
    }
    if (threadIdx.x == 0) partial[blockIdx.x] = sm[0];
}

__global__ __launch_bounds__(256) void reduce_max_stage2(
    const float* __restrict__ partial, int n, float* __restrict__ outc)
{
    __shared__ float sm[256];
    float m = -INFINITY;
    for (int i = threadIdx.x; i < n; i += 256) m = fmaxf(m, partial[i]);
    sm[threadIdx.x] = m;
    __syncthreads();
    for (int s = 128; s > 0; s >>= 1) {
        if (threadIdx.x < s) sm[threadIdx.x] = fmaxf(sm[threadIdx.x], sm[threadIdx.x + s]);
        __syncthreads();
    }
    if (threadIdx.x == 0) outc[0] = sm[0];
}

// ---------------- fused reweight kernel ----------------
// Block: 256 threads = 8 waves; wave handles 2 adjacent pixels in a row.
// Zero-padded LDS tiles encode the 2-pixel block-diagonal structure:
//   A tile 16x24 (rows 0-7: pixel0 heads, k in cols 0-8;
//                 rows 8-15: pixel1 heads, k in cols 12-20; rest zero)
//   B tile 24x16 (rows 0-8: P_p0[k][n]; rows 12-20: P_p1[k][n];
//                 rows 9-11,21-23 and cols 9-15 zero)
// D = A @ B via 6 unconditional V_WMMA_F32_16X16X4_F32 chunks (full f32).
#define AK 24          // padded K extent (6 chunks of 4)
#define AS 25          // A row stride (odd => conflict-free lane-indexed rows)

__global__ __launch_bounds__(256) void attn_reweight_kernel(
    const float* __restrict__ attn, const float* __restrict__ sims,
    const int* __restrict__ sinds, const float* __restrict__ cptr,
    float* __restrict__ out)
{
    __shared__ float sAz[8][16][AS];        // padded E tile (A matrix)
    __shared__ float sBz[8][AK][16];        // padded P tile (B matrix)
    __shared__ float sR [8][2][HD_][NSP_];  // 1/(eps + D)
    __shared__ int   sCI[8][2][NSP_];       // center sinds

    const int tid  = threadIdx.x;
    const int wave = tid >> 5;
    const int lane = tid & 31;

    // block -> (b, h, w0); W=192 = 12 tiles of 16 pixels
    const int blocksPerRow = W_ / 16;
    const int blocksPerImg = H_ * blocksPerRow;
    int bi = blockIdx.x;
    const int b = bi / blocksPerImg; bi -= b * blocksPerImg;
    const int h  = bi / blocksPerRow;
    const int w0 = (bi % blocksPerRow) * 16;
    const int wbase = w0 + wave * 2;           // this wave's pixel pair

    const float c = cptr[0];

    // ---- phase -1: zero the padded tiles (in-order LDS => fills may follow) ----
    {
        float* a = &sAz[wave][0][0];
        for (int i = lane; i < 16 * AS; i += 32) a[i] = 0.0f;
        float* bz = &sBz[wave][0][0];
        for (int i = lane; i < AK * 16; i += 32) bz[i] = 0.0f;
    }

    // ---- phase 0: center sinds ----
    if (lane < 2 * NSP_) {
        int p = lane / NSP_, n = lane % NSP_;
        sCI[wave][p][n] = sinds[(((size_t)b * H_ + h) * W_ + (wbase + p)) * NSP_ + n];
    }
    // ---- phase 1: E = exp(attn - c) into padded A tile ----
    for (int i = lane; i < 2 * HD_ * K_; i += 32) {
        int p  = i / (HD_ * K_);
        int r  = i - p * (HD_ * K_);
        int hd = r / K_, k = r % K_;
        float a = attn[((((size_t)b * HD_ + hd) * H_ + h) * W_ + (wbase + p)) * K_ + k];
        sAz[wave][p * 8 + hd][p * 12 + k] = expf(a - c);
    }
    __syncthreads();

    // ---- phase 2: P[p][k][n] into padded B tile ----
    if (lane < 2 * K_) {
        int p = lane / K_, k = lane % K_;
        int w = wbase + p;
        int nh = h + (k / 3) - 1; nh = nh < 0 ? 0 : (nh > H_ - 1 ? H_ - 1 : nh);
        int nw = w + (k % 3) - 1; nw = nw < 0 ? 0 : (nw > W_ - 1 ? W_ - 1 : nw);
        size_t base = (((size_t)b * H_ + nh) * W_ + nw) * NSP_;
        float acc[NSP_];
#pragma unroll
        for (int n = 0; n < NSP_; ++n) acc[n] = 0.0f;
#pragma unroll
        for (int s = 0; s < NSP_; ++s) {
            float sv = sims[base + s];
            int   iv = sinds[base + s];
#pragma unroll
            for (int n = 0; n < NSP_; ++n)
                acc[n] += (iv == sCI[wave][p][n]) ? sv : 0.0f;
        }
#pragma unroll
        for (int n = 0; n < NSP_; ++n) sBz[wave][p * 12 + k][n] = acc[n];
    }
    __syncthreads();

    // ---- phase 3: D = A @ B, 6 unconditional f32 WMMA chunks ----
    const int m     = lane & 15;            // A row / D row-base for this lane
    const int n16   = lane & 15;            // B/D column
    const int khalf = (lane >> 4) << 1;     // 0 or 2: which K pair this lane holds
    v8f acc = {};
#pragma unroll
    for (int ch = 0; ch < 6; ++ch) {
        const int k0 = 4 * ch + khalf;
        v2f av; v2f bv;
        av[0] = sAz[wave][m][k0];
        av[1] = sAz[wave][m][k0 + 1];
        bv[0] = sBz[wave][k0][n16];
        bv[1] = sBz[wave][k0 + 1][n16];
        acc = __builtin_amdgcn_wmma_f32_16x16x4_f32(
            false, av, false, bv, (short)0, acc, false, false);
    }
    // C/D layout: VGPR v, lanes 0-15 -> M=v, lanes 16-31 -> M=v+8
#pragma unroll
    for (int v = 0; v < 8; ++v) {
        int mr = (lane < 16) ? v : (v + 8);
        int p = mr >> 3, hd = mr & 7;
        if (n16 < NSP_) sR[wave][p][hd][n16] = 1.0f / (1e-10f + acc[v]);
    }
    __syncthreads();

    // ---- phase 4: streaming writes, k-contiguous (18 floats per (hd,n) per pair) ----
    for (int i = lane; i < 2 * HD_ * NSP_ * K_; i += 32) {
        int k = i % K_;
        int t = i / K_;
        int p = t & 1; t >>= 1;
        int n = t % NSP_;
        int hd = t / NSP_;
        float val = sAz[wave][p * 8 + hd][p * 12 + k]
                  * sBz[wave][p * 12 + k][n]
                  * sR[wave][p][hd][n];
        size_t oi = (((((size_t)b * HD_ + hd) * NSP_ + n) * H_ + h) * W_ + (wbase + p)) * K_ + k;
        __builtin_nontemporal_store(val, out + oi);
    }
}

extern "C" void kernel_launch(void* const* d_in, const int* in_sizes, int n_in,
                              void* d_out, int out_size, void* d_ws, size_t ws_size,
                              hipStream_t stream) {
    const float* attn  = (const float*)d_in[0];
    const float* sims  = (const float*)d_in[1];
    const int*   sinds = (const int*)d_in[2];
    float* out = (float*)d_out;
    float* ws  = (float*)d_ws;     // ws[0] = c, ws[1..512] = block partial maxes

    const int n_attn = in_sizes[0];
    reduce_max_stage1<<<512, 256, 0, stream>>>(attn, n_attn, ws + 1);
    reduce_max_stage2<<<1, 256, 0, stream>>>(ws + 1, 512, ws);

    const int nblocks = (B_ * H_ * W_) / 16;   // 4608 blocks, 16 pixels each
    attn_reweight_kernel<<<nblocks, 256, 0, stream>>>(attn, sims, sinds, ws, out);
}